// CenterLossA_51951924413098
// MI455X (gfx1250) — compile-verified
//
#include <hip/hip_runtime.h>

// ---------------------------------------------------------------------------
// Center-loss on MI455X (gfx1250).
// Pass 1: bandwidth-bound streaming reduction (256 MB of feat @ 23.3 TB/s
//         floor ~11 us). Pure VALU FMA chains, float4 loads, centers in LDS,
//         deterministic per-block partials (no float atomics).
// Pass 2: single-wave finalize; partial sums reduced on the matrix pipe via
//         chained V_WMMA_F32_16X16X4_F32 against an all-ones B matrix.
// ---------------------------------------------------------------------------

#define TPB      256          // 8 wave32s per block
#define NBLOCKS  1024         // 2*NBLOCKS floats of workspace (8 KB)
#define MAXD     2048         // FEAT_DIM (static LDS sizing)

typedef __attribute__((ext_vector_type(2))) float v2f;
typedef __attribute__((ext_vector_type(8))) float v8f;

__global__ __launch_bounds__(TPB) void center_loss_partials(
    const float* __restrict__ feat,
    const int*   __restrict__ label,
    const float* __restrict__ centers,
    float*       __restrict__ partials,   // [2 * gridDim.x]: own | all
    int B, int D)
{
    __shared__ __align__(16) float sC[3 * MAXD];
    __shared__ float sOwn[TPB];
    __shared__ float sAll[TPB];

    const int tid = threadIdx.x;

    // Stage centers (3*D floats = 24 KB) into LDS once per block.
    for (int i = tid; i < 3 * D; i += TPB) sC[i] = centers[i];
    __syncthreads();

    const int lane        = tid & 31;
    const int wave_in_blk = tid >> 5;
    const int wave_gid    = blockIdx.x * (TPB >> 5) + wave_in_blk;
    const int total_waves = gridDim.x * (TPB >> 5);
    const int cols4       = D >> 2;

    const float4* __restrict__ c0v = (const float4*)(sC);
    const float4* __restrict__ c1v = (const float4*)(sC + D);
    const float4* __restrict__ c2v = (const float4*)(sC + 2 * D);

    float own = 0.0f, all = 0.0f;

    // Each wave owns whole rows: label is wave-uniform -> scalar load/select.
    for (int r = wave_gid; r < B; r += total_waves) {
        const int l = label[r];
        const float4* __restrict__ frow =
            (const float4*)(feat + (size_t)r * (size_t)D);

        float a0 = 0.0f, a1 = 0.0f, a2 = 0.0f;

        // Wave reads 512 B contiguous per iteration (32 lanes x float4).
#pragma unroll 4
        for (int j = lane; j < cols4; j += 32) {
            const float4 f  = frow[j];
            const float4 c0 = c0v[j];
            const float4 c1 = c1v[j];
            const float4 c2 = c2v[j];

            float d;
            d = f.x - c0.x; a0 = fmaf(d, d, a0);
            d = f.y - c0.y; a0 = fmaf(d, d, a0);
            d = f.z - c0.z; a0 = fmaf(d, d, a0);
            d = f.w - c0.w; a0 = fmaf(d, d, a0);

            d = f.x - c1.x; a1 = fmaf(d, d, a1);
            d = f.y - c1.y; a1 = fmaf(d, d, a1);
            d = f.z - c1.z; a1 = fmaf(d, d, a1);
            d = f.w - c1.w; a1 = fmaf(d, d, a1);

            d = f.x - c2.x; a2 = fmaf(d, d, a2);
            d = f.y - c2.y; a2 = fmaf(d, d, a2);
            d = f.z - c2.z; a2 = fmaf(d, d, a2);
            d = f.w - c2.w; a2 = fmaf(d, d, a2);
        }

        own += (l == 0) ? a0 : ((l == 1) ? a1 : a2);
        all += a0 + a1 + a2;
    }

    // Deterministic block tree reduction.
    sOwn[tid] = own;
    sAll[tid] = all;
    __syncthreads();
    for (int s = TPB >> 1; s > 0; s >>= 1) {
        if (tid < s) {
            sOwn[tid] += sOwn[tid + s];
            sAll[tid] += sAll[tid + s];
        }
        __syncthreads();
    }
    if (tid == 0) {
        partials[blockIdx.x]             = sOwn[0];
        partials[gridDim.x + blockIdx.x] = sAll[0];
    }
}

// Single-wave finalize: reduce nPart partials per stream on the matrix pipe.
// B = all-ones => D[m,n] = sum_k A[m,k] + C[m,n]; accumulate over chunks of 64.
__global__ __launch_bounds__(32) void center_loss_finalize(
    const float* __restrict__ partials, float* __restrict__ out,
    int nPart, float invB)
{
    const int lane = threadIdx.x;          // wave32, EXEC all ones for WMMA
    const int half = lane >> 4;            // 0: K=0,1  1: K=2,3
    const int lm   = lane & 15;

    v2f bones = {1.0f, 1.0f};              // B[k,n] = 1 for all k,n
    v8f accO  = {};
    v8f accA  = {};

    for (int chunk = 0; chunk < nPart; chunk += 64) {
        const int idx = chunk + half * 32 + 2 * lm;   // covers chunk..chunk+63
        v2f aO = { partials[idx],         partials[idx + 1] };
        v2f aA = { partials[nPart + idx], partials[nPart + idx + 1] };

        accO = __builtin_amdgcn_wmma_f32_16x16x4_f32(
                   false, aO, false, bones, (short)0, accO, false, false);
        accA = __builtin_amdgcn_wmma_f32_16x16x4_f32(
                   false, aA, false, bones, (short)0, accA, false, false);
    }

    // Column n=0 lives in lane 0 (rows m=0..7 across the 8 acc VGPRs) and
    // lane 16 (rows m=8..15).
    float so = 0.0f, sa = 0.0f;
#pragma unroll
    for (int v = 0; v < 8; ++v) { so += accO[v]; sa += accA[v]; }

    const float mainSum = __shfl(so, 0, 32) + __shfl(so, 16, 32);
    const float allSum  = __shfl(sa, 0, 32) + __shfl(sa, 16, 32);

    if (lane == 0) {
        const float distocen = allSum - mainSum;   // sum over the two "other" centers
        out[0] = mainSum * (1.0f + 1.0f / distocen) * 0.5f * invB;
    }
}

extern "C" void kernel_launch(void* const* d_in, const int* in_sizes, int n_in,
                              void* d_out, int out_size, void* d_ws, size_t ws_size,
                              hipStream_t stream) {
    const float* feat    = (const float*)d_in[0];
    const int*   label   = (const int*)d_in[1];
    const float* centers = (const float*)d_in[2];
    float*       out     = (float*)d_out;

    const int B = in_sizes[1];                 // 32768
    const int D = in_sizes[0] / B;             // 2048

    // Partial-count must be a multiple of 64 for the WMMA finalize.
    int nblocks = NBLOCKS;
    while ((size_t)(2 * nblocks) * sizeof(float) > ws_size && nblocks > 64)
        nblocks >>= 1;

    float* partials = (float*)d_ws;

    center_loss_partials<<<nblocks, TPB, 0, stream>>>(
        feat, label, centers, partials, B, D);
    center_loss_finalize<<<1, 32, 0, stream>>>(
        partials, out, nblocks, 1.0f / (float)B);
}